// CountHistogram_33809982554604
// MI455X (gfx1250) — compile-verified
//
#include <hip/hip_runtime.h>
#include <cstdint>

#define NBINS 29
#define B_    64
#define C_    4
#define Q_    32
#define D_    4096
#define HALF_ (D_ / 2)   // 2048 elements per wave

// One 256-thread block (8 wave32s) per (b,q) pair.
//  - mask row (4096 B) staged into LDS with CDNA5 async copy (ASYNCcnt path)
//  - simmat streamed with coalesced b128 loads + global_prefetch
//  - per-wave privatized histograms in LDS, accumulated with ds_add_f32
//  - final cross-wave reduction writes 4*29 floats, each exactly once
__global__ __launch_bounds__(256, 4)
void CountHistogram_33809982554604_kernel(const float* __restrict__ simmat,
                                          const unsigned char* __restrict__ mask,
                                          float* __restrict__ out)
{
    __shared__ unsigned char smask[D_];    // staged mask row, 4 KB
    __shared__ float shist[8 * 33];        // 8 waves x 29 bins, padded to 33 dwords

    const int tid = threadIdx.x;
    const int wg  = blockIdx.x;            // [0, B*Q)
    const int b   = wg >> 5;               // / Q_
    const int q   = wg & (Q_ - 1);         // % Q_

    // ---- stage the (b,q) mask row into LDS via CDNA5 async-to-LDS copy ----
    // 256 lanes x 16 B = 4096 B; one global_load_async_to_lds_b128 per wave.
    {
        const unsigned char* g = mask + ((size_t)b * Q_ + q) * D_ + tid * 16;
        uint32_t lds_off = (uint32_t)(uintptr_t)(&smask[tid * 16]);
        asm volatile("global_load_async_to_lds_b128 %0, %1, off"
                     :: "v"(lds_off), "v"(g)
                     : "memory");
    }

    // zero per-wave histograms while the async copy is in flight
    for (int i = tid; i < 8 * 33; i += 256) shist[i] = 0.0f;

#if __has_builtin(__builtin_amdgcn_s_wait_asynccnt)
    __builtin_amdgcn_s_wait_asynccnt(0);
#else
    asm volatile("s_wait_asynccnt 0" ::: "memory");
#endif
    __syncthreads();

    const int lane = tid & 31;
    const int w    = tid >> 5;    // wave id 0..7
    const int c    = w >> 1;      // channel 0..3
    const int h    = w & 1;       // which half of the D dimension

    const float* srow =
        simmat + (((size_t)(b * C_ + c) * Q_ + q) * D_) + (size_t)h * HALF_;
    float* hw = &shist[w * 33];
    const unsigned char* mhalf = &smask[h * HALF_];

#pragma unroll 4
    for (int it = 0; it < 16; ++it) {
        const int off = it * 128 + lane * 4;         // coalesced: 512 B per wave op
        // prefetch ~2 KB ahead in the stream (gfx1250 global_prefetch_b8)
        __builtin_prefetch(srow + off + 512, 0, 1);

        const float4   s  = *(const float4*)(srow + off);
        const uint32_t m4 = *(const uint32_t*)(mhalf + off);  // 4 mask bytes (LDS)

        // bin = trunc(((s + 1.00001)/2) * 28); inputs in [-1,1) => bin in [0,28]
        if (m4 & 0x000000FFu) {
            int bin = (int)((s.x + 1.00001f) * 0.5f * 28.0f);
            __hip_atomic_fetch_add(&hw[bin], 1.0f, __ATOMIC_RELAXED,
                                   __HIP_MEMORY_SCOPE_WORKGROUP);
        }
        if (m4 & 0x0000FF00u) {
            int bin = (int)((s.y + 1.00001f) * 0.5f * 28.0f);
            __hip_atomic_fetch_add(&hw[bin], 1.0f, __ATOMIC_RELAXED,
                                   __HIP_MEMORY_SCOPE_WORKGROUP);
        }
        if (m4 & 0x00FF0000u) {
            int bin = (int)((s.z + 1.00001f) * 0.5f * 28.0f);
            __hip_atomic_fetch_add(&hw[bin], 1.0f, __ATOMIC_RELAXED,
                                   __HIP_MEMORY_SCOPE_WORKGROUP);
        }
        if (m4 & 0xFF000000u) {
            int bin = (int)((s.w + 1.00001f) * 0.5f * 28.0f);
            __hip_atomic_fetch_add(&hw[bin], 1.0f, __ATOMIC_RELAXED,
                                   __HIP_MEMORY_SCOPE_WORKGROUP);
        }
    }

    __syncthreads();

    // ---- reduce the two waves per channel and store (each output once) ----
    if (tid < C_ * NBINS) {
        const int c2 = tid / NBINS;
        const int n  = tid - c2 * NBINS;
        const float v = shist[(2 * c2) * 33 + n] + shist[(2 * c2 + 1) * 33 + n];
        out[(((size_t)b * C_ + c2) * Q_ + q) * NBINS + n] = v;
    }
}

extern "C" void kernel_launch(void* const* d_in, const int* in_sizes, int n_in,
                              void* d_out, int out_size, void* d_ws, size_t ws_size,
                              hipStream_t stream) {
    (void)in_sizes; (void)n_in; (void)out_size; (void)d_ws; (void)ws_size;
    const float*         simmat = (const float*)d_in[0];
    const unsigned char* mask   = (const unsigned char*)d_in[1];  // jax bool: 1 byte/elem
    float*               out    = (float*)d_out;

    CountHistogram_33809982554604_kernel<<<dim3(B_ * Q_), dim3(256), 0, stream>>>(
        simmat, mask, out);
}